// kmeans_cluster_4612794876725
// MI455X (gfx1250) — compile-verified
//
#include <hip/hip_runtime.h>
#include <math.h>

#define KC   128      // clusters
#define DD   768      // feature dim
#define BB   8192     // batch
#define LRATE 0.001f
#define EPSN  1e-8f

typedef __attribute__((ext_vector_type(2))) float v2f;
typedef __attribute__((ext_vector_type(8))) float v8f;
typedef __attribute__((ext_vector_type(2))) float f2;
typedef __attribute__((ext_vector_type(4))) float f4;
typedef __attribute__((ext_vector_type(4))) int   i4;

// ---------------------------------------------------------------------------
// Normalize centroids and repack into (k-pair, column)-major float2 buffer:
// bufB[pair p][col k] = { cn[k][2p], cn[k][2p+1] }  -> one b64 load per lane
// in the WMMA B-operand build.
// ---------------------------------------------------------------------------
__global__ void centroid_pack(const float* __restrict__ cent, float* __restrict__ bufB) {
    __shared__ float red[512];
    const int k = blockIdx.x;      // 0..127
    const int t = threadIdx.x;     // 0..383 (one float2 each)
    f2 v = ((const f2*)(cent + (size_t)k * DD))[t];
    red[t] = v.x * v.x + v.y * v.y;
    if (t < 128) red[384 + t] = 0.0f;
    __syncthreads();
    for (int off = 256; off > 0; off >>= 1) {
        if (t < off) red[t] += red[t + off];
        __syncthreads();
    }
    const float inv = 1.0f / fmaxf(sqrtf(red[0]), EPSN);
    f2 o; o.x = v.x * inv; o.y = v.y * inv;
    ((f2*)bufB)[(size_t)t * KC + k] = o;
}

// ---------------------------------------------------------------------------
// Per-datapoint inverse L2 norm (reference: x / max(||x||, EPS))
// ---------------------------------------------------------------------------
__global__ void dp_norms(const float* __restrict__ dp, float* __restrict__ invn) {
    __shared__ float red[512];
    const int b = blockIdx.x;
    const int t = threadIdx.x;     // 0..383
    f2 v = ((const f2*)(dp + (size_t)b * DD))[t];
    red[t] = v.x * v.x + v.y * v.y;
    if (t < 128) red[384 + t] = 0.0f;
    __syncthreads();
    for (int off = 256; off > 0; off >>= 1) {
        if (t < off) red[t] += red[t + off];
        __syncthreads();
    }
    if (t == 0) invn[b] = 1.0f / fmaxf(sqrtf(red[0]), EPSN);
}

// ---------------------------------------------------------------------------
// Cosine-similarity GEMM via V_WMMA_F32_16X16X4_F32 + fused argmax.
// One wave32 per 16-row M-tile; 8 accumulators cover all 128 centroids.
// A layout (16x4 f32): lanes 0-15 M=lane K={0,1}; lanes 16-31 M=lane-16 K={2,3}
// C layout: VGPR j -> row j (lanes 0-15) / j+8 (lanes 16-31), col = lane&15.
// ---------------------------------------------------------------------------
__global__ void cossim_wmma(const float* __restrict__ dp,
                            const float* __restrict__ bufB,
                            const float* __restrict__ invn,
                            float* __restrict__ intra,
                            int* __restrict__ dpIdx) {
    const int wave  = threadIdx.x >> 5;
    const int lane  = threadIdx.x & 31;
    const int mtile = blockIdx.x * 8 + wave;
    const int half  = lane >> 4;
    const int col0  = lane & 15;
    const int row   = mtile * 16 + col0;

    const float ainv = invn[row];
    const f2* __restrict__ A2 = (const f2*)(dp + (size_t)row * DD);
    const f2* __restrict__ B2 = (const f2*)bufB;

    v8f acc[8] = {};

    for (int k0 = 0; k0 < DD; k0 += 4) {
        const int kp = (k0 >> 1) + half;         // k-pair index for this lane
        f2 av = A2[kp];
        v2f a; a.x = av.x * ainv; a.y = av.y * ainv;
#pragma unroll
        for (int n = 0; n < 8; ++n) {
            f2 bv = B2[(size_t)kp * KC + n * 16 + col0];
            v2f b; b.x = bv.x; b.y = bv.y;
            acc[n] = __builtin_amdgcn_wmma_f32_16x16x4_f32(
                false, a, false, b, (short)0, acc[n], false, false);
        }
    }

    // Store intra_cos_sim tile + per-row argmax (lowest-index tie-break).
#pragma unroll
    for (int j = 0; j < 8; ++j) {
        const int r = mtile * 16 + j + 8 * half;
        float bv = -1.0e30f;
        int   bc = 0;
#pragma unroll
        for (int n = 0; n < 8; ++n) {
            const float v = acc[n][j];
            intra[(size_t)r * KC + n * 16 + col0] = v;
            const int c = n * 16 + col0;
            if (v > bv || (v == bv && c < bc)) { bv = v; bc = c; }
        }
#pragma unroll
        for (int m = 1; m < 16; m <<= 1) {   // reduce within 16-lane half
            const float ov = __shfl_xor(bv, m, 32);
            const int   oc = __shfl_xor(bc, m, 32);
            if (ov > bv || (ov == bv && oc < bc)) { bv = ov; bc = oc; }
        }
        if (col0 == 0) dpIdx[r] = bc;
    }
}

// ---------------------------------------------------------------------------
// dp_cluster[b][c] = (idx[b]==idx[c]) && b!=c  — 268 MB of streaming stores.
// ---------------------------------------------------------------------------
__global__ void cluster_fill(const int* __restrict__ idx, float* __restrict__ out2) {
    const int b  = blockIdx.y;
    const int c4 = blockIdx.x * blockDim.x + threadIdx.x;   // quad index
    const int kb = idx[b];
    const i4  kc = ((const i4*)idx)[c4];
    const int c0 = c4 * 4;
    f4 o;
    o.x = (kc.x == kb && (c0 + 0) != b) ? 1.0f : 0.0f;
    o.y = (kc.y == kb && (c0 + 1) != b) ? 1.0f : 0.0f;
    o.z = (kc.z == kb && (c0 + 2) != b) ? 1.0f : 0.0f;
    o.w = (kc.w == kb && (c0 + 3) != b) ? 1.0f : 0.0f;
    __builtin_nontemporal_store(o, (f4*)(out2 + (size_t)b * BB + c0));
}

// ---------------------------------------------------------------------------
// dp_centroid[b] = centroid[idx[b]]  (centroid rows stay L2-resident)
// ---------------------------------------------------------------------------
__global__ void gather_centroid(const int* __restrict__ idx,
                                const float* __restrict__ cent,
                                float* __restrict__ out3) {
    const int b = blockIdx.x;
    const int t = threadIdx.x;     // 0..191 (one float4 each)
    const int k = idx[b];
    f4 v = ((const f4*)(cent + (size_t)k * DD))[t];
    __builtin_nontemporal_store(v, (f4*)(out3 + (size_t)b * DD) + t);
}

__global__ void zero_ws(float* __restrict__ p, int n) {
    const int i = blockIdx.x * blockDim.x + threadIdx.x;
    if (i < n) p[i] = 0.0f;
}

// ---------------------------------------------------------------------------
// Scatter-sum raw datapoints into per-cluster accumulators (L2-resident).
// ---------------------------------------------------------------------------
__global__ void scatter_add(const float* __restrict__ dp, const int* __restrict__ idx,
                            float* __restrict__ acc, float* __restrict__ counts) {
    const int b = blockIdx.x;
    const int t = threadIdx.x;     // 0..255
    const int k = idx[b];
    const float* __restrict__ row = dp + (size_t)b * DD;
    float* __restrict__ arow = acc + (size_t)k * DD;
    atomicAdd(arow + t,       row[t]);
    atomicAdd(arow + t + 256, row[t + 256]);
    atomicAdd(arow + t + 512, row[t + 512]);
    if (t == 0) atomicAdd(counts + k, 1.0f);
}

// new_centroid = LR*(acc + centroid*(count==0)) + (1-LR)*centroid
__global__ void finalize_centroid(const float* __restrict__ acc,
                                  const float* __restrict__ counts,
                                  const float* __restrict__ cent,
                                  float* __restrict__ out4) {
    const int e = blockIdx.x * blockDim.x + threadIdx.x;
    if (e >= KC * DD) return;
    const int k = e / DD;
    const float base = cent[e];
    const float data = acc[e] + ((counts[k] > 0.0f) ? 0.0f : base);
    out4[e] = LRATE * data + (1.0f - LRATE) * base;
}

extern "C" void kernel_launch(void* const* d_in, const int* in_sizes, int n_in,
                              void* d_out, int out_size, void* d_ws, size_t ws_size,
                              hipStream_t stream) {
    const float* dp   = (const float*)d_in[0];   // [B, D]
    const float* cent = (const float*)d_in[2];   // [K, D]  (d_in[1] is unused batch_cos_sim)

    float* out1 = (float*)d_out;                                   // intra_cos_sim [B,K]
    float* out2 = out1 + (size_t)BB * KC;                          // dp_cluster   [B,B]
    float* out3 = out2 + (size_t)BB * BB;                          // dp_centroid  [B,D]
    float* out4 = out3 + (size_t)BB * DD;                          // new_centroid [K,D]

    char* w = (char*)d_ws;
    float* bufB   = (float*)w;  w += sizeof(float) * KC * DD;      // packed normalized centroids
    float* invn   = (float*)w;  w += sizeof(float) * BB;           // 1/||dp_b||
    int*   dpIdx  = (int*)w;    w += sizeof(int)   * BB;           // argmax indices
    float* acc    = (float*)w;  w += sizeof(float) * KC * DD;      // cluster sums
    float* counts = (float*)w;  w += sizeof(float) * KC;           // cluster counts (contiguous after acc)

    centroid_pack<<<KC, 384, 0, stream>>>(cent, bufB);
    dp_norms<<<BB, 384, 0, stream>>>(dp, invn);
    zero_ws<<<(KC * DD + KC + 255) / 256, 256, 0, stream>>>(acc, KC * DD + KC);

    cossim_wmma<<<BB / 16 / 8, 256, 0, stream>>>(dp, bufB, invn, out1, dpIdx);

    cluster_fill<<<dim3(BB / (256 * 4), BB), 256, 0, stream>>>(dpIdx, out2);
    gather_centroid<<<BB, 192, 0, stream>>>(dpIdx, cent, out3);
    scatter_add<<<BB, 256, 0, stream>>>(dp, dpIdx, acc, counts);
    finalize_centroid<<<(KC * DD + 255) / 256, 256, 0, stream>>>(acc, counts, cent, out4);
}